// SINE_60043642798860
// MI455X (gfx1250) — compile-verified
//
#include <hip/hip_runtime.h>
#include <math.h>

// ---------------- problem constants (match reference setup_inputs) ----------
#define B_    256
#define T_CNT 200
#define TP    208      // T padded to 13 * 16 for WMMA M-tiles
#define D_    128
#define LDA   132      // padded LDS row stride (floats) to dodge bank conflicts
#define L_    500
#define K_    4
#define NT_M  13
#define NT_N  8

static constexpr int SM_FLOATS =
    2 * TP * LDA      // s_x, s_y
  + 2 * TP            // s_h, s_a
  + 2 * K_ * TP       // s_hk (P_tk), s_P (P_ktb)
  + 512               // s_su
  + D_                // s_zu
  + 3 * K_ * D_       // s_Cu, s_Cun, s_delta
  + D_                // s_Capt
  + 256               // s_red
  + 16                // s_ek(8) + s_topv(8)
  + TP + 8;           // s_seq (int) + s_topi (int)

typedef float v2f __attribute__((ext_vector_type(2)));
typedef float v8f __attribute__((ext_vector_type(8)));

// fp32 WMMA: D(16x16) = A(16x4) * B(4x16) + C   -> v_wmma_f32_16x16x4_f32
__device__ __forceinline__ v8f wmma4(v2f a, v2f b, v8f c) {
  return __builtin_amdgcn_wmma_f32_16x16x4_f32(false, a, false, b, (short)0, c,
                                               false, false);
}

// ---------------- reduction helpers -----------------------------------------
__device__ __forceinline__ float wave_allreduce_sum(float v) {
  v += __shfl_xor(v, 16, 32);
  v += __shfl_xor(v, 8, 32);
  v += __shfl_xor(v, 4, 32);
  v += __shfl_xor(v, 2, 32);
  v += __shfl_xor(v, 1, 32);
  return v;
}

__device__ __forceinline__ float block_reduce_sum(float v, float* red, int tid) {
  red[tid] = v;
  __syncthreads();
  #pragma unroll
  for (int s = 128; s >= 1; s >>= 1) {
    if (tid < s) red[tid] += red[tid + s];
    __syncthreads();
  }
  float r = red[0];
  __syncthreads();
  return r;
}

__device__ __forceinline__ float block_reduce_max(float v, float* red, int tid) {
  red[tid] = v;
  __syncthreads();
  #pragma unroll
  for (int s = 128; s >= 1; s >>= 1) {
    if (tid < s) red[tid] = fmaxf(red[tid], red[tid + s]);
    __syncthreads();
  }
  float r = red[0];
  __syncthreads();
  return r;
}

// softmax over t=0..T_CNT-1 of h[], result into dst[] (dst may alias h)
__device__ __forceinline__ void softmax_T(const float* h, float* dst, float* red,
                                          int tid) {
  float v = (tid < T_CNT) ? h[tid] : -3.0e38f;
  float mx = block_reduce_max(v, red, tid);
  float e = (tid < T_CNT) ? __expf(v - mx) : 0.f;
  float s = block_reduce_sum(e, red, tid);
  if (tid < T_CNT) dst[tid] = e / s;
  __syncthreads();
}

// ---------------- WMMA tile workers ------------------------------------------
// y = As(TPxD_, LDS, stride LDA) @ W(D_xD_, global row-major)
// variant 1: hout[m] += sum_n tanh(y[m,n]) * wv[n]   (hout pre-zeroed LDS)
__device__ __forceinline__ void mm_tanh_dot_accum(const float* __restrict__ As,
                                                  const float* __restrict__ W,
                                                  const float* __restrict__ wv,
                                                  float* __restrict__ hout,
                                                  int lane, int wave) {
  const int half = lane >> 4;      // 0: K{0,1}, 1: K{2,3}
  const int l15 = lane & 15;
  for (int tile = wave; tile < NT_M * NT_N; tile += 8) {
    const int m0 = (tile >> 3) << 4;
    const int n0 = (tile & 7) << 4;
    v8f acc = {};
    const float* arow = As + (m0 + l15) * LDA + 2 * half;
    const float* bptr = W + (2 * half) * D_ + n0 + l15;
    #pragma unroll 4
    for (int k0 = 0; k0 < D_; k0 += 4) {
      v2f a, b;
      a.x = arow[k0];
      a.y = arow[k0 + 1];
      b.x = bptr[k0 * D_];
      b.y = bptr[k0 * D_ + D_];
      acc = wmma4(a, b, acc);
    }
    const float wl = wv[n0 + l15];
    #pragma unroll
    for (int r = 0; r < 8; ++r) {
      float v = tanhf(acc[r]) * wl;
      v += __shfl_xor(v, 8, 32);   // reduce across the 16 columns held by the
      v += __shfl_xor(v, 4, 32);   // half-wave (bit 4 untouched -> halves stay
      v += __shfl_xor(v, 2, 32);   // independent)
      v += __shfl_xor(v, 1, 32);
      if (l15 == 0) atomicAdd(&hout[m0 + (half << 3) + r], v);  // ds_add_f32
    }
  }
}

// variant 2: store full product into Ys (LDS, stride LDA)
__device__ __forceinline__ void mm_store(const float* __restrict__ As,
                                         const float* __restrict__ W,
                                         float* __restrict__ Ys,
                                         int lane, int wave) {
  const int half = lane >> 4;
  const int l15 = lane & 15;
  for (int tile = wave; tile < NT_M * NT_N; tile += 8) {
    const int m0 = (tile >> 3) << 4;
    const int n0 = (tile & 7) << 4;
    v8f acc = {};
    const float* arow = As + (m0 + l15) * LDA + 2 * half;
    const float* bptr = W + (2 * half) * D_ + n0 + l15;
    #pragma unroll 4
    for (int k0 = 0; k0 < D_; k0 += 4) {
      v2f a, b;
      a.x = arow[k0];
      a.y = arow[k0 + 1];
      b.x = bptr[k0 * D_];
      b.y = bptr[k0 * D_ + D_];
      acc = wmma4(a, b, acc);
    }
    #pragma unroll
    for (int r = 0; r < 8; ++r)
      Ys[(m0 + (half << 3) + r) * LDA + n0 + l15] = acc[r];
  }
}

// ---------------- main kernel: one workgroup per batch row -------------------
__global__ __launch_bounds__(256) void sine_kernel(
    const int* __restrict__ item_seq, const float* __restrict__ item_emb,
    const float* __restrict__ Cm, const float* __restrict__ w1,
    const float* __restrict__ w2, const float* __restrict__ w3,
    const float* __restrict__ w4, const float* __restrict__ wk1,
    const float* __restrict__ wk2, const float* __restrict__ g2,
    const float* __restrict__ b2, const float* __restrict__ g4,
    const float* __restrict__ b4, float* __restrict__ out) {
  extern __shared__ float sm[];
  float* s_x = sm;                   // TP*LDA : x_u (zero-padded rows T..207)
  float* s_y = s_x + TP * LDA;       // TP*LDA : x_u@w3, later x_u_bar
  float* s_h = s_y + TP * LDA;       // TP     : dot-reduce accumulator
  float* s_a = s_h + TP;             // TP     : attention weights
  float* s_hk = s_a + TP;            // K_*TP  : per-k accumulator -> P_tk
  float* s_P = s_hk + K_ * TP;       // TP*K_  : P_ktb[t][k]
  float* s_su = s_P + TP * K_;       // 512    : s_u scores
  float* s_zu = s_su + 512;          // D_
  float* s_Cu = s_zu + D_;           // K_*D_
  float* s_Cun = s_Cu + K_ * D_;     // K_*D_
  float* s_delta = s_Cun + K_ * D_;  // K_*D_
  float* s_Capt = s_delta + K_ * D_; // D_
  float* s_red = s_Capt + D_;        // 256
  float* s_ek = s_red + 256;         // 8
  float* s_topv = s_ek + 8;          // 8
  int* s_seq = (int*)(s_topv + 8);   // TP
  int* s_topi = s_seq + TP;          // 8

  const int tid = threadIdx.x;
  const int lane = tid & 31;
  const int wave = tid >> 5;
  const int b = blockIdx.x;

  // ---- 1) stage sequence ids, gather x_u (float4), zero accumulators -------
  for (int i = tid; i < TP; i += 256) {
    s_seq[i] = (i < T_CNT) ? item_seq[b * T_CNT + i] : 0;
    s_h[i] = 0.f;
  }
  for (int i = tid; i < K_ * TP; i += 256) s_hk[i] = 0.f;
  __syncthreads();
  for (int idx = tid; idx < TP * (D_ / 4); idx += 256) {
    const int t = idx >> 5;
    const int q = idx & 31;
    float4 v = make_float4(0.f, 0.f, 0.f, 0.f);
    if (t < T_CNT)
      v = ((const float4*)(item_emb + (size_t)s_seq[t] * D_))[q];
    ((float4*)(s_x + t * LDA))[q] = v;
  }
  __syncthreads();

  // ---- 2) a = softmax_t( tanh(x_u@w1) @ w2 ) --------------------------------
  mm_tanh_dot_accum(s_x, w1, w2, s_h, lane, wave);
  __syncthreads();
  softmax_T(s_h, s_a, s_red, tid);

  // ---- 3) z_u = sum_t a[t] x_u[t,:] -----------------------------------------
  if (tid < D_) {
    float acc = 0.f;
    for (int t = 0; t < T_CNT; ++t) acc += s_a[t] * s_x[t * LDA + tid];
    s_zu[tid] = acc;
  }
  __syncthreads();

  // ---- 4) s_u = z_u @ C^T ---------------------------------------------------
  for (int l = tid; l < L_; l += 256) {
    const float* cr = Cm + (size_t)l * D_;
    float acc = 0.f;
    for (int d = 0; d < D_; ++d) acc += s_zu[d] * cr[d];
    s_su[l] = acc;
  }
  __syncthreads();

  // ---- 5) top-K (then reversed -> ascending, as in reference) ---------------
  if (tid == 0) {
    float tv[K_];
    int ti[K_];
    for (int k = 0; k < K_; ++k) {
      float best = -3.0e38f;
      int bi = 0;
      for (int l = 0; l < L_; ++l)
        if (s_su[l] > best) { best = s_su[l]; bi = l; }
      tv[k] = best;
      ti[k] = bi;
      s_su[bi] = -3.0e38f;
    }
    for (int k = 0; k < K_; ++k) {  // reverse
      s_topv[k] = tv[K_ - 1 - k];
      s_topi[k] = ti[K_ - 1 - k];
    }
  }
  __syncthreads();

  // ---- 6) C_u = C[idx]*sigmoid(s), C_u_n = LN(C_u; g2,b2) — one wave per k --
  if (wave < K_) {
    const int k = wave;
    const int id = s_topi[k];
    const float sg = 1.f / (1.f + __expf(-s_topv[k]));
    float c[4];
    float sum = 0.f;
    #pragma unroll
    for (int j = 0; j < 4; ++j) {
      const int d = lane + 32 * j;
      c[j] = Cm[(size_t)id * D_ + d] * sg;
      s_Cu[k * D_ + d] = c[j];
      sum += c[j];
    }
    sum = wave_allreduce_sum(sum);
    const float mean = sum * (1.f / D_);
    float ss = 0.f;
    #pragma unroll
    for (int j = 0; j < 4; ++j) {
      const float dvv = c[j] - mean;
      ss += dvv * dvv;
    }
    ss = wave_allreduce_sum(ss);
    const float rstd = rsqrtf(ss * (1.f / D_) + 1e-12f);
    #pragma unroll
    for (int j = 0; j < 4; ++j) {
      const int d = lane + 32 * j;
      s_Cun[k * D_ + d] = (c[j] - mean) * rstd * g2[d] + b2[d];
    }
  }
  __syncthreads();

  // ---- 7) y3 = x_u @ w3 (kept in LDS) ---------------------------------------
  mm_store(s_x, w3, s_y, lane, wave);
  __syncthreads();

  // ---- 8) P_ktb[t,:] = softmax_k( l2norm(y3[t]) . C_u_n[k] ) ----------------
  if (tid < T_CNT) {
    const float* yr = s_y + tid * LDA;
    float ss = 0.f;
    for (int d = 0; d < D_; ++d) ss += yr[d] * yr[d];
    const float inv = 1.f / fmaxf(sqrtf(ss), 1e-12f);
    float sc[K_];
    float mx = -3.0e38f;
    for (int k = 0; k < K_; ++k) {
      float dot = 0.f;
      const float* cn = s_Cun + k * D_;
      for (int d = 0; d < D_; ++d) dot += yr[d] * cn[d];
      sc[k] = dot * inv;
      mx = fmaxf(mx, sc[k]);
    }
    float se = 0.f;
    for (int k = 0; k < K_; ++k) {
      sc[k] = __expf(sc[k] - mx);
      se += sc[k];
    }
    for (int k = 0; k < K_; ++k) s_P[tid * K_ + k] = sc[k] / se;
  }
  __syncthreads();

  // ---- 9) P_tk[k,:] = softmax_t( tanh(x_u@wk1[k]) @ wk2[k] ) ----------------
  for (int k = 0; k < K_; ++k)
    mm_tanh_dot_accum(s_x, wk1 + (size_t)k * D_ * D_, wk2 + k * D_,
                      s_hk + k * TP, lane, wave);
  __syncthreads();
  for (int k = 0; k < K_; ++k)
    softmax_T(s_hk + k * TP, s_hk + k * TP, s_red, tid);

  // ---- 10) delta_k = l2norm( sum_t P_ktb[t,k]*P_tk[k,t]*x_u[t,:] ) ----------
  for (int idx = tid; idx < K_ * D_; idx += 256) {
    const int k = idx >> 7;
    const int d = idx & (D_ - 1);
    float acc = 0.f;
    for (int t = 0; t < T_CNT; ++t)
      acc += s_P[t * K_ + k] * s_hk[k * TP + t] * s_x[t * LDA + d];
    s_delta[idx] = acc;
  }
  __syncthreads();
  if (wave < K_) {
    const int k = wave;
    float v[4];
    float ss = 0.f;
    #pragma unroll
    for (int j = 0; j < 4; ++j) {
      v[j] = s_delta[k * D_ + lane + 32 * j];
      ss += v[j] * v[j];
    }
    ss = wave_allreduce_sum(ss);
    const float inv = 1.f / fmaxf(sqrtf(ss), 1e-12f);
    #pragma unroll
    for (int j = 0; j < 4; ++j) s_delta[k * D_ + lane + 32 * j] = v[j] * inv;
  }
  __syncthreads();

  // ---- 11) x_u_bar = P_ktb @ C_u  (overwrites s_y; pad rows zeroed) ---------
  for (int idx = tid; idx < TP * D_; idx += 256) {
    const int t = idx >> 7;
    const int d = idx & (D_ - 1);
    float v = 0.f;
    if (t < T_CNT) {
      #pragma unroll
      for (int k = 0; k < K_; ++k) v += s_P[t * K_ + k] * s_Cu[k * D_ + d];
    }
    s_y[t * LDA + d] = v;
  }
  for (int i = tid; i < TP; i += 256) s_h[i] = 0.f;
  __syncthreads();

  // ---- 12) apt_w = softmax_t( tanh(x_u_bar@w3) @ w4 ) -----------------------
  mm_tanh_dot_accum(s_y, w3, w4, s_h, lane, wave);
  __syncthreads();
  softmax_T(s_h, s_a, s_red, tid);

  // ---- 13) C_apt = LN( sum_t apt_w[t] x_u_bar[t,:]; g4,b4 ) -----------------
  float capt = 0.f;
  if (tid < D_)
    for (int t = 0; t < T_CNT; ++t) capt += s_a[t] * s_y[t * LDA + tid];
  const float mean = block_reduce_sum((tid < D_) ? capt : 0.f, s_red, tid) *
                     (1.f / D_);
  const float dv = (tid < D_) ? (capt - mean) : 0.f;
  const float var = block_reduce_sum(dv * dv, s_red, tid) * (1.f / D_);
  if (tid < D_) s_Capt[tid] = dv * rsqrtf(var + 1e-12f) * g4[tid] + b4[tid];
  __syncthreads();

  // ---- 14) e_ku = softmax_k( delta_k . C_apt / TAU );  v_u = e @ delta ------
  if (wave < K_) {
    const int k = wave;
    float dot = 0.f;
    #pragma unroll
    for (int j = 0; j < 4; ++j) {
      const int d = lane + 32 * j;
      dot += s_delta[k * D_ + d] * s_Capt[d];
    }
    dot = wave_allreduce_sum(dot);
    if (lane == 0) s_ek[k] = dot * 10.0f;  // 1/TAU
  }
  __syncthreads();
  if (tid == 0) {
    float mx = fmaxf(fmaxf(s_ek[0], s_ek[1]), fmaxf(s_ek[2], s_ek[3]));
    float e[K_];
    float s = 0.f;
    for (int k = 0; k < K_; ++k) {
      e[k] = __expf(s_ek[k] - mx);
      s += e[k];
    }
    for (int k = 0; k < K_; ++k) s_ek[k] = e[k] / s;
  }
  __syncthreads();
  if (tid < D_) {
    float v = 0.f;
    #pragma unroll
    for (int k = 0; k < K_; ++k) v += s_ek[k] * s_delta[k * D_ + tid];
    out[(size_t)b * D_ + tid] = v;
  }
}

// ---------------- item_emb passthrough (second output) -----------------------
__global__ __launch_bounds__(256) void copy_emb(const float4* __restrict__ src,
                                                float4* __restrict__ dst,
                                                int n4) {
  int i = blockIdx.x * blockDim.x + threadIdx.x;
  const int stride = gridDim.x * blockDim.x;
  for (; i < n4; i += stride) dst[i] = src[i];
}

// ---------------- host entry -------------------------------------------------
extern "C" void kernel_launch(void* const* d_in, const int* in_sizes, int n_in,
                              void* d_out, int out_size, void* d_ws,
                              size_t ws_size, hipStream_t stream) {
  (void)in_sizes; (void)n_in; (void)out_size; (void)d_ws; (void)ws_size;
  const int* item_seq = (const int*)d_in[0];
  const float* item_emb = (const float*)d_in[1];
  const float* Cm = (const float*)d_in[2];
  const float* w1 = (const float*)d_in[3];
  const float* w2 = (const float*)d_in[4];
  const float* w3 = (const float*)d_in[5];
  const float* w4 = (const float*)d_in[6];
  const float* wk1 = (const float*)d_in[7];
  const float* wk2 = (const float*)d_in[8];
  const float* g2 = (const float*)d_in[9];
  const float* b2 = (const float*)d_in[10];
  const float* g4 = (const float*)d_in[11];
  const float* b4 = (const float*)d_in[12];
  float* out = (float*)d_out;

  const size_t smem = (size_t)SM_FLOATS * sizeof(float);  // ~239 KB < 320 KB/WGP
  sine_kernel<<<dim3(B_), dim3(256), smem, stream>>>(
      item_seq, item_emb, Cm, w1, w2, w3, w4, wk1, wk2, g2, b2, g4, b4, out);

  const int n4 = (100000 * D_) / 4;
  copy_emb<<<dim3((n4 + 255) / 256), dim3(256), 0, stream>>>(
      (const float4*)item_emb, (float4*)(out + (size_t)B_ * D_), n4);
}